// NumEncoder_43533788512746
// MI455X (gfx1250) — compile-verified
//
#include <hip/hip_runtime.h>
#include <math.h>
#include <stdint.h>

typedef float v2f __attribute__((ext_vector_type(2)));
typedef float v4f __attribute__((ext_vector_type(4)));
typedef float v8f __attribute__((ext_vector_type(8)));

#define WMMA_F32(acc, a, b)                                                  \
  acc = __builtin_amdgcn_wmma_f32_16x16x4_f32(false, a, false, b, (short)0,  \
                                              acc, false, false)

// ---------------------------------------------------------------- constants
// B=32, S=4096, N=64, D=512, HOPS=2
#define BB 32
#define SS 4096
#define NN 64
#define DD 512

// ---- CDNA5 async global->LDS copy (16B per lane), tracked by ASYNCcnt ----
__device__ __forceinline__ void async_ld_b128(void* lds_ptr, const void* gptr) {
  const uint32_t l = (uint32_t)(uintptr_t)lds_ptr;      // low 32 bits = LDS offset
  const uint64_t g = (uint64_t)(uintptr_t)gptr;
  asm volatile("global_load_async_to_lds_b128 %0, %1, off"
               :: "v"(l), "v"(g) : "memory");
}
__device__ __forceinline__ void wait_async0() {
  asm volatile("s_wait_asynccnt 0" ::: "memory");
}

// ---------------------------------------------------------------- graphs
__global__ __launch_bounds__(64) void k_graph(const int* __restrict__ order,
                                              float* __restrict__ g1,
                                              float* __restrict__ g2) {
  const int b = blockIdx.x;
  const int t = threadIdx.x;  // 64 threads, one row each
  __shared__ int so[NN];
  so[t] = order[b * NN + t];
  __syncthreads();
  const int oi = so[t];
  const float mi = (oi > 0) ? 1.f : 0.f;
  float s1 = 0.f, s2 = 0.f;
  for (int j = 0; j < NN; ++j) {
    const int oj = so[j];
    const float mj = (oj > 0) ? 1.f : 0.f;
    const float e = (j == t) ? 1.f : 0.f;
    const float base = mi * mj * (1.f - e);
    const float gr = (oi > oj) ? 1.f : 0.f;
    s1 += base * gr + e;
    s2 += base * (1.f - gr) + e;
  }
  const float inv1 = 1.f / s1, inv2 = 1.f / s2;
  float* r1 = g1 + (b * NN + t) * NN;
  float* r2 = g2 + (b * NN + t) * NN;
  for (int j = 0; j < NN; ++j) {
    const int oj = so[j];
    const float mj = (oj > 0) ? 1.f : 0.f;
    const float e = (j == t) ? 1.f : 0.f;
    const float base = mi * mj * (1.f - e);
    const float gr = (oi > oj) ? 1.f : 0.f;
    r1[j] = (base * gr + e) * inv1;
    r2[j] = (base * (1.f - gr) + e) * inv2;
  }
}

// ------------------------------------------------- m1/m2 = g1/g2 @ node (K=64)
__global__ __launch_bounds__(256) void k_gemm_graph(
    const float* __restrict__ g1, const float* __restrict__ g2,
    const float* __restrict__ node, float* __restrict__ m1,
    float* __restrict__ m2) {
  __shared__ float sG[2][NN][68];
  __shared__ float sN[NN][68];
  const int b = blockIdx.y;
  const int c0 = blockIdx.x * 64;
  const int tid = threadIdx.x;
  const float* gsrc0 = g1 + b * NN * NN;
  const float* gsrc1 = g2 + b * NN * NN;
  for (int idx = tid * 4; idx < NN * NN; idx += 1024) {
    const int r = idx >> 6, c = idx & 63;
    async_ld_b128(&sG[0][r][c], gsrc0 + idx);
    async_ld_b128(&sG[1][r][c], gsrc1 + idx);
    async_ld_b128(&sN[r][c], node + (b * NN + r) * DD + c0 + c);
  }
  wait_async0();
  __syncthreads();
  const int w = tid >> 5, lane = tid & 31;
  const int sel = w >> 2, wm = w & 3;
  const int laneM = lane & 15, laneHi = lane >> 4;
  v8f acc0 = {}, acc1 = {}, acc2 = {}, acc3 = {};
  for (int k = 0; k < NN; k += 4) {
    v2f a;
    a.x = sG[sel][16 * wm + laneM][k + 2 * laneHi];
    a.y = sG[sel][16 * wm + laneM][k + 2 * laneHi + 1];
    v2f b0, b1, b2, b3;
    b0.x = sN[k + 2 * laneHi][laneM];
    b0.y = sN[k + 2 * laneHi + 1][laneM];
    b1.x = sN[k + 2 * laneHi][16 + laneM];
    b1.y = sN[k + 2 * laneHi + 1][16 + laneM];
    b2.x = sN[k + 2 * laneHi][32 + laneM];
    b2.y = sN[k + 2 * laneHi + 1][32 + laneM];
    b3.x = sN[k + 2 * laneHi][48 + laneM];
    b3.y = sN[k + 2 * laneHi + 1][48 + laneM];
    WMMA_F32(acc0, a, b0);
    WMMA_F32(acc1, a, b1);
    WMMA_F32(acc2, a, b2);
    WMMA_F32(acc3, a, b3);
  }
  float* mo = (sel == 0 ? m1 : m2) + b * NN * DD;
#pragma unroll
  for (int v = 0; v < 8; ++v) {
    const int row = 16 * wm + 8 * laneHi + v;
    mo[row * DD + c0 + laneM] = acc0[v];
    mo[row * DD + c0 + 16 + laneM] = acc1[v];
    mo[row * DD + c0 + 32 + laneM] = acc2[v];
    mo[row * DD + c0 + 48 + laneM] = acc3[v];
  }
}

// -------------------------------------------- C = relu(A[2048,512] @ W[512,512])
__global__ __launch_bounds__(256) void k_gemm_relu(const float* __restrict__ A,
                                                   const float* __restrict__ Wt,
                                                   float* __restrict__ C) {
  __shared__ float sA[64][36];
  __shared__ float sW[32][68];
  const int r0 = blockIdx.y * 64;
  const int c0 = blockIdx.x * 64;
  const int tid = threadIdx.x;
  const int w = tid >> 5, lane = tid & 31;
  const int wm = w & 3, wn = w >> 2;
  const int laneM = lane & 15, laneHi = lane >> 4;
  v8f acc0 = {}, acc1 = {};
  for (int k0 = 0; k0 < DD; k0 += 32) {
    __syncthreads();
    {
      const int idx = tid * 4;
      const int r = idx >> 5, kk = idx & 31;
      async_ld_b128(&sA[r][kk], A + (r0 + r) * DD + k0 + kk);
      async_ld_b128(&sA[r + 32][kk], A + (r0 + r + 32) * DD + k0 + kk);
      const int rw = idx >> 6, cw = idx & 63;
      async_ld_b128(&sW[rw][cw], Wt + (k0 + rw) * DD + c0 + cw);
      async_ld_b128(&sW[rw + 16][cw], Wt + (k0 + rw + 16) * DD + c0 + cw);
    }
    wait_async0();
    __syncthreads();
#pragma unroll
    for (int kk = 0; kk < 32; kk += 4) {
      v2f a;
      a.x = sA[16 * wm + laneM][kk + 2 * laneHi];
      a.y = sA[16 * wm + laneM][kk + 2 * laneHi + 1];
      v2f b0, b1;
      b0.x = sW[kk + 2 * laneHi][32 * wn + laneM];
      b0.y = sW[kk + 2 * laneHi + 1][32 * wn + laneM];
      b1.x = sW[kk + 2 * laneHi][32 * wn + 16 + laneM];
      b1.y = sW[kk + 2 * laneHi + 1][32 * wn + 16 + laneM];
      WMMA_F32(acc0, a, b0);
      WMMA_F32(acc1, a, b1);
    }
  }
#pragma unroll
  for (int v = 0; v < 8; ++v) {
    const int row = r0 + 16 * wm + 8 * laneHi + v;
    const int col = c0 + 32 * wn + laneM;
    C[row * DD + col] = fmaxf(acc0[v], 0.f);
    C[row * DD + col + 16] = fmaxf(acc1[v], 0.f);
  }
}

// ---- T = node + sigmoid(I1@W1)*I1 + sigmoid(I2@W2)*I2  (dual GEMM fused)
__global__ __launch_bounds__(256) void k_gate(
    const float* __restrict__ I1, const float* __restrict__ I2,
    const float* __restrict__ W1, const float* __restrict__ W2,
    const float* __restrict__ node, float* __restrict__ T) {
  __shared__ float sA1[64][36], sA2[64][36];
  __shared__ float sW1[32][68], sW2[32][68];
  const int r0 = blockIdx.y * 64;
  const int c0 = blockIdx.x * 64;
  const int tid = threadIdx.x;
  const int w = tid >> 5, lane = tid & 31;
  const int wm = w & 3, wn = w >> 2;
  const int laneM = lane & 15, laneHi = lane >> 4;
  v8f a10 = {}, a11 = {}, a20 = {}, a21 = {};
  for (int k0 = 0; k0 < DD; k0 += 32) {
    __syncthreads();
    {
      const int idx = tid * 4;
      const int r = idx >> 5, kk = idx & 31;
      async_ld_b128(&sA1[r][kk], I1 + (r0 + r) * DD + k0 + kk);
      async_ld_b128(&sA1[r + 32][kk], I1 + (r0 + r + 32) * DD + k0 + kk);
      async_ld_b128(&sA2[r][kk], I2 + (r0 + r) * DD + k0 + kk);
      async_ld_b128(&sA2[r + 32][kk], I2 + (r0 + r + 32) * DD + k0 + kk);
      const int rw = idx >> 6, cw = idx & 63;
      async_ld_b128(&sW1[rw][cw], W1 + (k0 + rw) * DD + c0 + cw);
      async_ld_b128(&sW1[rw + 16][cw], W1 + (k0 + rw + 16) * DD + c0 + cw);
      async_ld_b128(&sW2[rw][cw], W2 + (k0 + rw) * DD + c0 + cw);
      async_ld_b128(&sW2[rw + 16][cw], W2 + (k0 + rw + 16) * DD + c0 + cw);
    }
    wait_async0();
    __syncthreads();
#pragma unroll
    for (int kk = 0; kk < 32; kk += 4) {
      v2f a, b0, b1;
      a.x = sA1[16 * wm + laneM][kk + 2 * laneHi];
      a.y = sA1[16 * wm + laneM][kk + 2 * laneHi + 1];
      b0.x = sW1[kk + 2 * laneHi][32 * wn + laneM];
      b0.y = sW1[kk + 2 * laneHi + 1][32 * wn + laneM];
      b1.x = sW1[kk + 2 * laneHi][32 * wn + 16 + laneM];
      b1.y = sW1[kk + 2 * laneHi + 1][32 * wn + 16 + laneM];
      WMMA_F32(a10, a, b0);
      WMMA_F32(a11, a, b1);
      a.x = sA2[16 * wm + laneM][kk + 2 * laneHi];
      a.y = sA2[16 * wm + laneM][kk + 2 * laneHi + 1];
      b0.x = sW2[kk + 2 * laneHi][32 * wn + laneM];
      b0.y = sW2[kk + 2 * laneHi + 1][32 * wn + laneM];
      b1.x = sW2[kk + 2 * laneHi][32 * wn + 16 + laneM];
      b1.y = sW2[kk + 2 * laneHi + 1][32 * wn + 16 + laneM];
      WMMA_F32(a20, a, b0);
      WMMA_F32(a21, a, b1);
    }
  }
#pragma unroll
  for (int v = 0; v < 8; ++v) {
    const int row = r0 + 16 * wm + 8 * laneHi + v;
    const int col = c0 + 32 * wn + laneM;
    {
      const float g1v = 1.f / (1.f + __expf(-a10[v]));
      const float g2v = 1.f / (1.f + __expf(-a20[v]));
      const float i1 = I1[row * DD + col];
      const float i2 = I2[row * DD + col];
      T[row * DD + col] = node[row * DD + col] + g1v * i1 + g2v * i2;
    }
    {
      const float g1v = 1.f / (1.f + __expf(-a11[v]));
      const float g2v = 1.f / (1.f + __expf(-a21[v]));
      const float i1 = I1[row * DD + col + 16];
      const float i2 = I2[row * DD + col + 16];
      T[row * DD + col + 16] =
          node[row * DD + col + 16] + g1v * i1 + g2v * i2;
    }
  }
}

// ---------------------------------------------------------------- layernorm
__global__ __launch_bounds__(128) void k_ln(const float* __restrict__ T,
                                            const float* __restrict__ scale,
                                            const float* __restrict__ bias,
                                            float* __restrict__ O) {
  const int row = blockIdx.x;  // 2048 rows
  const int t = threadIdx.x;   // 128 threads * float4 = 512
  const float4 v = *(const float4*)(T + row * DD + t * 4);
  float s = v.x + v.y + v.z + v.w;
  float sq = v.x * v.x + v.y * v.y + v.z * v.z + v.w * v.w;
#pragma unroll
  for (int off = 16; off > 0; off >>= 1) {
    s += __shfl_xor(s, off, 32);
    sq += __shfl_xor(sq, off, 32);
  }
  __shared__ float ps[4], pq[4];
  const int wv = t >> 5, lane = t & 31;
  if (lane == 0) { ps[wv] = s; pq[wv] = sq; }
  __syncthreads();
  s = ps[0] + ps[1] + ps[2] + ps[3];
  sq = pq[0] + pq[1] + pq[2] + pq[3];
  const float mean = s * (1.f / 512.f);
  const float var = sq * (1.f / 512.f) - mean * mean;
  const float inv = rsqrtf(var + 1e-5f);
  const float4 sc = *(const float4*)(scale + t * 4);
  const float4 bi = *(const float4*)(bias + t * 4);
  float4 o;
  o.x = (v.x - mean) * inv * sc.x + bi.x;
  o.y = (v.y - mean) * inv * sc.y + bi.y;
  o.z = (v.z - mean) * inv * sc.z + bi.z;
  o.w = (v.w - mean) * inv * sc.w + bi.w;
  *(float4*)(O + row * DD + t * 4) = o;
}

// ---------------------------------------------------------------- elementwise
__global__ void k_copy(const float* __restrict__ src, float* __restrict__ dst,
                       int n) {
  const int i = (blockIdx.x * blockDim.x + threadIdx.x) * 4;
  if (i < n) *(float4*)(dst + i) = *(const float4*)(src + i);
}

__global__ void k_add(const float* __restrict__ a, const float* __restrict__ b,
                      float* __restrict__ dst, int n) {
  const int i = (blockIdx.x * blockDim.x + threadIdx.x) * 4;
  if (i < n) {
    const float4 x = *(const float4*)(a + i);
    const float4 y = *(const float4*)(b + i);
    float4 o;
    o.x = x.x + y.x; o.y = x.y + y.y; o.z = x.z + y.z; o.w = x.w + y.w;
    *(float4*)(dst + i) = o;
  }
}

// -------- fused: out1 = encoder + scatter(node); partial column maxima
// encoder/out1 are touched exactly once (268MB each) -> non-temporal;
// node (4MB) is reused by every s-chunk -> keep cacheable.
__global__ __launch_bounds__(128) void k_stream(
    const float* __restrict__ enc, const float* __restrict__ node,
    const int* __restrict__ pos, float* __restrict__ out1,
    float* __restrict__ partial) {
  const int b = blockIdx.y;
  const int s0 = blockIdx.x * 128;
  const int t = threadIdx.x;
  __shared__ int map[128];
  map[t] = -1;
  __syncthreads();
  if (t < NN) {
    const int p = pos[b * NN + t];
    if (p >= s0 && p < s0 + 128) map[p - s0] = t;
  }
  __syncthreads();
  const int d = t * 4;
  v4f mx = {-3.4e38f, -3.4e38f, -3.4e38f, -3.4e38f};
  for (int ss = 0; ss < 128; ++ss) {
    const size_t off = (((size_t)(s0 + ss)) * BB + b) * DD + d;
    v4f v = __builtin_nontemporal_load((const v4f*)(enc + off));
    const int n = map[ss];
    if (n >= 0) {
      const v4f nv = *(const v4f*)(node + (b * NN + n) * DD + d);
      v += nv;
    }
    __builtin_nontemporal_store(v, (v4f*)(out1 + off));
    mx.x = fmaxf(mx.x, v.x);
    mx.y = fmaxf(mx.y, v.y);
    mx.z = fmaxf(mx.z, v.z);
    mx.w = fmaxf(mx.w, v.w);
  }
  *(v4f*)(partial + ((size_t)blockIdx.x * BB + b) * DD + d) = mx;
}

__global__ void k_maxred(const float* __restrict__ partial,
                         float* __restrict__ out3) {
  const int idx = (blockIdx.x * blockDim.x + threadIdx.x) * 4;  // over B*D
  if (idx >= BB * DD) return;
  float4 m = *(const float4*)(partial + idx);
  for (int c = 1; c < 32; ++c) {
    const float4 v = *(const float4*)(partial + c * (BB * DD) + idx);
    m.x = fmaxf(m.x, v.x);
    m.y = fmaxf(m.y, v.y);
    m.z = fmaxf(m.z, v.z);
    m.w = fmaxf(m.w, v.w);
  }
  *(float4*)(out3 + idx) = m;
}

// ---------------------------------------------------------------- launcher
extern "C" void kernel_launch(void* const* d_in, const int* in_sizes, int n_in,
                              void* d_out, int out_size, void* d_ws,
                              size_t ws_size, hipStream_t stream) {
  (void)in_sizes; (void)n_in; (void)out_size; (void)ws_size;
  const float* enc = (const float*)d_in[0];     // [S,B,D]
  const float* numenc = (const float*)d_in[1];  // [B,N,D]
  const float* w1a = (const float*)d_in[2];     // [HOPS,D,D]
  const float* w1b = (const float*)d_in[3];
  const float* w2a = (const float*)d_in[4];
  const float* w2b = (const float*)d_in[5];
  const float* lns = (const float*)d_in[6];     // [HOPS,D]
  const float* lnb = (const float*)d_in[7];
  const int* pos = (const int*)d_in[8];         // [B,N]
  const int* ord = (const int*)d_in[9];         // [B,N]

  float* out1 = (float*)d_out;                       // gnn_info_vec [S,B,D]
  float* out2 = out1 + (size_t)SS * BB * DD;         // num_embedding [B,N,D]
  float* out3 = out2 + (size_t)BB * NN * DD;         // problem_output [B,D]

  const int NODE = BB * NN * DD;  // 1048576
  float* wsf = (float*)d_ws;
  float* nodeA = wsf;
  float* nodeB = nodeA + NODE;
  float* g1 = nodeB + NODE;
  float* g2 = g1 + BB * NN * NN;
  float* m1 = g2 + BB * NN * NN;
  float* m2 = m1 + NODE;
  float* i1 = m2 + NODE;
  float* i2 = i1 + NODE;
  float* part = i2 + NODE;  // 32*32*512

  k_graph<<<dim3(BB), dim3(64), 0, stream>>>(ord, g1, g2);
  k_copy<<<dim3(NODE / (256 * 4)), dim3(256), 0, stream>>>(numenc, nodeA, NODE);

  float* cur = nodeA;
  float* nxt = nodeB;
  for (int h = 0; h < 2; ++h) {
    k_gemm_graph<<<dim3(8, BB), dim3(256), 0, stream>>>(g1, g2, cur, m1, m2);
    k_gemm_relu<<<dim3(8, 32), dim3(256), 0, stream>>>(m1, w1a + h * DD * DD, i1);
    k_gemm_relu<<<dim3(8, 32), dim3(256), 0, stream>>>(m2, w2a + h * DD * DD, i2);
    k_gate<<<dim3(8, 32), dim3(256), 0, stream>>>(i1, i2, w1b + h * DD * DD,
                                                  w2b + h * DD * DD, cur, m1);
    k_ln<<<dim3(BB * NN), dim3(128), 0, stream>>>(m1, lns + h * DD,
                                                  lnb + h * DD, nxt);
    float* tmp = cur; cur = nxt; nxt = tmp;
  }

  k_add<<<dim3(NODE / (256 * 4)), dim3(256), 0, stream>>>(numenc, cur, out2, NODE);
  k_stream<<<dim3(SS / 128, BB), dim3(128), 0, stream>>>(enc, cur, pos, out1, part);
  k_maxred<<<dim3(16), dim3(256), 0, stream>>>(part, out3);
}